// SparseAutoencoderAuxLoss_66812511256587
// MI455X (gfx1250) — compile-verified
//
#include <hip/hip_runtime.h>

typedef float v2f __attribute__((ext_vector_type(2)));
typedef float v8f __attribute__((ext_vector_type(8)));
typedef unsigned int u32x4 __attribute__((ext_vector_type(4)));
typedef int i32x4 __attribute__((ext_vector_type(4)));
typedef int i32x8 __attribute__((ext_vector_type(8)));

#define BATCH 4096
#define DIN   2048
#define HID   16384
#define KTOP  64

// ---------------- encoder GEMM: h_raw = x @ W_enc^T + b_enc ----------------
// Block tile 128x128, 256 threads (8 waves, wave32). Wave tile 64x32 =
// 4x2 subtiles of 16x16, K stepped by 4 via V_WMMA_F32_16X16X4_F32.
// Staging: double-buffered Tensor Data Mover loads (tensor_load_to_lds),
// with hardware LDS padding 16DW data + 4DW pad = 80B row stride (LDT=20),
// which is both 16B-aligned and LDS-bank-conflict-free for the fragment reads.
#define BM  128
#define BN  128
#define BKE 16
#define LDT 20

__device__ __forceinline__ unsigned int flipKey(float f) {
  unsigned int u = __float_as_uint(f);
  return (u & 0x80000000u) ? ~u : (u | 0x80000000u);
}

// Issue one TDM 2-D tile load: tileRows x 16 f32 from row-major matrix
// (row stride DIN) into LDS with 4-DWORD padding after every 16 DWORDs.
__device__ __forceinline__ void tdm_load_tile(const float* gptr, void* lptr,
                                              int tileRows) {
  const unsigned long long ga = (unsigned long long)(uintptr_t)gptr;
  const unsigned int la = (unsigned int)(uintptr_t)lptr;  // low 32b of flat LDS ptr = LDS offset
  u32x4 g0;
  g0[0] = 1u;                                          // count=1 (valid user D#)
  g0[1] = la;                                          // lds_addr
  g0[2] = (unsigned int)(ga & 0xFFFFFFFFull);          // global_addr[31:0]
  g0[3] = (unsigned int)((ga >> 32) & 0x01FFFFFFull)   // global_addr[56:32]
        | (2u << 30);                                  // type = 2 ("image")
  i32x8 g1;
  g1[0] = (2 << 16)      // data_size = 4 bytes
        | (1 << 20)      // pad_enable
        | (3 << 22)      // pad_interval: 16 DWORDs between pads
        | (3 << 25);     // pad_amount: 4 DWORDs of pad
  g1[1] = (int)((DIN & 0xFFFF) << 16);                 // tensor_dim0[15:0] @ [63:48]
  g1[2] = (int)(((DIN >> 16) & 0xFFFF) |               // tensor_dim0[31:16]
                ((HID & 0xFFFF) << 16));               // tensor_dim1[15:0]
  g1[3] = (int)(((HID >> 16) & 0xFFFF) |               // tensor_dim1[31:16]
                (BKE << 16));                          // tile_dim0 = 16
  g1[4] = tileRows;                                    // tile_dim1; tile_dim2 = 0
  g1[5] = DIN;                                         // tensor_dim0_stride[31:0]
  g1[6] = 0;                                           // stride hi / dim1_stride (unused, 2-D)
  g1[7] = 0;
  const i32x4 z4 = {0, 0, 0, 0};                       // D# groups 2/3: dims 2..4 unused
  const i32x8 z8 = {0, 0, 0, 0, 0, 0, 0, 0};
  __builtin_amdgcn_tensor_load_to_lds(g0, g1, z4, z4, z8, 0);
}

__global__ __launch_bounds__(256) void sae_encoder_wmma(
    const float* __restrict__ X, const float* __restrict__ We,
    const float* __restrict__ be, float* __restrict__ Hout)
{
  __shared__ float As[2][BM * LDT];
  __shared__ float Bs[2][BN * LDT];

  const int tid  = threadIdx.x;
  const int lane = tid & 31;
  const int wave = tid >> 5;     // 0..7
  const int half = lane >> 4;    // 0/1 -> K sub-pair
  const int l16  = lane & 15;    // M (A) / N (B,C,D)
  const int wm   = wave >> 2;    // 0..1 : 64-row strip
  const int wn   = wave & 3;     // 0..3 : 32-col strip
  const int bm0  = blockIdx.y * BM;
  const int bn0  = blockIdx.x * BN;

  const int nIter = DIN / BKE;   // 128

  // prologue: DMA first K-tile pair into buffer 0 (wave 0 issues; TDM ignores EXEC)
  if (wave == 0) {
    tdm_load_tile(&X[(size_t)bm0 * DIN], &As[0][0], BM);
    tdm_load_tile(&We[(size_t)bn0 * DIN], &Bs[0][0], BN);
  }

  v8f acc[4][2];
#pragma unroll
  for (int m = 0; m < 4; ++m)
#pragma unroll
    for (int n = 0; n < 2; ++n)
      acc[m][n] = (v8f){};

  for (int it = 0; it < nIter; ++it) {
    const int cur = it & 1;
    if (wave == 0) {
      if (it + 1 < nIter) {
        const int k1 = (it + 1) * BKE;
        tdm_load_tile(&X[(size_t)bm0 * DIN + k1], &As[cur ^ 1][0], BM);
        tdm_load_tile(&We[(size_t)bn0 * DIN + k1], &Bs[cur ^ 1][0], BN);
        // in-order TDM: <=2 outstanding means the current buffer's pair landed
        __builtin_amdgcn_s_wait_tensorcnt(2);
      } else {
        __builtin_amdgcn_s_wait_tensorcnt(0);
      }
    }
    __syncthreads();

    const float* Ab = &As[cur][0];
    const float* Bb = &Bs[cur][0];
#pragma unroll
    for (int kk = 0; kk < BKE; kk += 4) {
      const int kf = kk + 2 * half;  // lanes 0-15: K={kk,kk+1}; lanes 16-31: K={kk+2,kk+3}
      v2f a[4], b[2];
#pragma unroll
      for (int m = 0; m < 4; ++m)
        a[m] = *(const v2f*)&Ab[(wm * 64 + m * 16 + l16) * LDT + kf];
#pragma unroll
      for (int n = 0; n < 2; ++n)
        b[n] = *(const v2f*)&Bb[(wn * 32 + n * 16 + l16) * LDT + kf];
#pragma unroll
      for (int m = 0; m < 4; ++m)
#pragma unroll
        for (int n = 0; n < 2; ++n)
          acc[m][n] = __builtin_amdgcn_wmma_f32_16x16x4_f32(
              false, a[m], false, b[n], (short)0, acc[m][n], false, false);
    }
    __syncthreads();
  }

  // epilogue: + b_enc, store. D layout: VGPR j -> M = j + 8*half, lane l16 -> N.
#pragma unroll
  for (int n = 0; n < 2; ++n) {
    const int gcol = bn0 + wn * 32 + n * 16 + l16;
    const float bias = be[gcol];
#pragma unroll
    for (int m = 0; m < 4; ++m) {
      const int grow0 = bm0 + wm * 64 + m * 16 + half * 8;
#pragma unroll
      for (int j = 0; j < 8; ++j)
        Hout[(size_t)(grow0 + j) * HID + gcol] = acc[m][n][j] + bias;
    }
  }
}

// ---------------- exact per-row top-64 via radix select, deterministic mask ---
__global__ __launch_bounds__(256) void sae_topk_mask(
    float* __restrict__ Hbuf, int* __restrict__ tkIdx, float* __restrict__ tkVal)
{
  const int row = blockIdx.x;
  float* hrow = Hbuf + (size_t)row * HID;
  const int tid = threadIdx.x;

  __shared__ unsigned int hist[256];
  __shared__ unsigned int s_prefix;
  __shared__ int s_r;
  __shared__ int s_waveEq[8], s_waveKp[8];
  __shared__ int s_eqBase, s_kpBase;

  if (tid == 0) { s_prefix = 0u; s_r = KTOP; }
  __syncthreads();

  // 4 passes of 8-bit radix select for the 64th-largest key
  for (int pass = 0; pass < 4; ++pass) {
    const int shift = 24 - pass * 8;
    const unsigned long long pref = (unsigned long long)s_prefix;
    hist[tid] = 0u;
    __syncthreads();
    for (int s = 0; s < HID / 256; ++s) {
      const unsigned int key = flipKey(hrow[s * 256 + tid]);
      if (((unsigned long long)key >> (shift + 8)) == pref)
        atomicAdd(&hist[(key >> shift) & 255u], 1u);
    }
    __syncthreads();
    if (tid == 0) {
      const int r = s_r;
      unsigned int cum = 0;
      int d = 255;
      for (; d >= 0; --d) { cum += hist[d]; if ((int)cum >= r) break; }
      s_r = r - (int)(cum - hist[d]);          // rank among threshold-equal keys
      s_prefix = (s_prefix << 8) | (unsigned int)d;
    }
    __syncthreads();
  }

  const unsigned int thrKey = s_prefix;  // exact key of 64th-largest element
  const int allowedEq = s_r;             // how many threshold-equal elems to keep
  if (tid == 0) { s_eqBase = 0; s_kpBase = 0; }
  __syncthreads();

  const int wv = tid >> 5, ln = tid & 31;
  const unsigned int ltmask = (1u << ln) - 1u;

  for (int c = 0; c < HID / 256; ++c) {
    const int i = c * 256 + tid;
    const float v = hrow[i];
    const unsigned int key = flipKey(v);
    const bool greater = key > thrKey;
    const bool equal   = key == thrKey;

    const unsigned int balEq = (unsigned int)__ballot(equal);
    if (ln == 0) s_waveEq[wv] = __popc(balEq);
    __syncthreads();
    int eqOff = s_eqBase;
    for (int w = 0; w < wv; ++w) eqOff += s_waveEq[w];
    const bool keepEq = equal && (eqOff + __popc(balEq & ltmask) < allowedEq);
    const bool keep = greater || keepEq;

    const unsigned int balK = (unsigned int)__ballot(keep);
    if (ln == 0) s_waveKp[wv] = __popc(balK);
    __syncthreads();
    int kpOff = s_kpBase;
    for (int w = 0; w < wv; ++w) kpOff += s_waveKp[w];
    if (keep) {
      const int slot = kpOff + __popc(balK & ltmask);  // index-ordered, deterministic
      tkIdx[(size_t)row * KTOP + slot] = i;
      tkVal[(size_t)row * KTOP + slot] = v;
    }
    hrow[i] = keep ? v : 0.0f;
    __syncthreads();
    if (tid == 0) {
      int e = 0, k = 0;
      for (int w = 0; w < 8; ++w) { e += s_waveEq[w]; k += s_waveKp[w]; }
      s_eqBase += e; s_kpBase += k;
    }
    __syncthreads();
  }
}

// ---------------- W_dec [D,H] -> Wt [H,D] so feature columns are contiguous ---
__global__ __launch_bounds__(256) void sae_transpose_wdec(
    const float* __restrict__ Wd, float* __restrict__ Wt)
{
  __shared__ float t[32][33];
  const int h0 = blockIdx.x * 32;
  const int d0 = blockIdx.y * 32;
  const int tx = threadIdx.x & 31, ty = threadIdx.x >> 5;
#pragma unroll
  for (int i = 0; i < 32; i += 8)
    t[ty + i][tx] = Wd[(size_t)(d0 + ty + i) * HID + h0 + tx];
  __syncthreads();
#pragma unroll
  for (int i = 0; i < 32; i += 8)
    Wt[(size_t)(h0 + ty + i) * DIN + d0 + tx] = t[tx][ty + i];
}

// ---------------- sparse decoder: x_hat[b,:] = b_dec + sum_j v_j * Wt[idx_j,:] --
__global__ __launch_bounds__(256) void sae_decoder(
    const float* __restrict__ Wt, const float* __restrict__ bd,
    const int* __restrict__ tkIdx, const float* __restrict__ tkVal,
    float* __restrict__ Xhat)
{
  const int row = blockIdx.x;
  const int tid = threadIdx.x;
  __shared__ int   sIdx[KTOP];
  __shared__ float sVal[KTOP];
  if (tid < KTOP) {
    sIdx[tid] = tkIdx[(size_t)row * KTOP + tid];
    sVal[tid] = tkVal[(size_t)row * KTOP + tid];
  }
  __syncthreads();

  float4 acc[2];
#pragma unroll
  for (int s = 0; s < 2; ++s)
    acc[s] = *(const float4*)&bd[(tid + s * 256) * 4];

  for (int j = 0; j < KTOP; ++j) {
    const float a = sVal[j];
    const float* wrow = Wt + (size_t)sIdx[j] * DIN;
#pragma unroll
    for (int s = 0; s < 2; ++s) {
      const float4 w = *(const float4*)&wrow[(tid + s * 256) * 4];
      acc[s].x += a * w.x; acc[s].y += a * w.y;
      acc[s].z += a * w.z; acc[s].w += a * w.w;
    }
  }
#pragma unroll
  for (int s = 0; s < 2; ++s)
    *(float4*)&Xhat[(size_t)row * DIN + (tid + s * 256) * 4] = acc[s];
}

extern "C" void kernel_launch(void* const* d_in, const int* in_sizes, int n_in,
                              void* d_out, int out_size, void* d_ws, size_t ws_size,
                              hipStream_t stream)
{
  (void)in_sizes; (void)n_in; (void)out_size; (void)ws_size;
  const float* x  = (const float*)d_in[0];   // [B, D]
  const float* We = (const float*)d_in[1];   // [H, D]
  const float* be = (const float*)d_in[2];   // [H]
  const float* Wd = (const float*)d_in[3];   // [D, H]
  const float* bd = (const float*)d_in[4];   // [D]

  float* h    = (float*)d_out;                        // [B, H]
  float* xhat = (float*)d_out + (size_t)BATCH * HID;  // [B, D]

  // workspace: Wt (H*D f32 = 128 MiB) | tkIdx (B*64 i32) | tkVal (B*64 f32)
  float* Wt    = (float*)d_ws;
  int*   tkIdx = (int*)((char*)d_ws + (size_t)HID * DIN * sizeof(float));
  float* tkVal = (float*)(tkIdx + (size_t)BATCH * KTOP);

  sae_transpose_wdec<<<dim3(HID / 32, DIN / 32), 256, 0, stream>>>(Wd, Wt);
  sae_encoder_wmma<<<dim3(HID / BN, BATCH / BM), 256, 0, stream>>>(x, We, be, h);
  sae_topk_mask<<<BATCH, 256, 0, stream>>>(h, tkIdx, tkVal);
  sae_decoder<<<BATCH, 256, 0, stream>>>(Wt, bd, tkIdx, tkVal, xhat);
}